// BondAwareEGNNLayer_50491635531849
// MI455X (gfx1250) — compile-verified
//
#include <hip/hip_runtime.h>
#include <hip/hip_bf16.h>

// ---------------------------------------------------------------------------
// BondAwareEGNNLayer for MI455X (gfx1250, wave32).
// fp32 WMMA (v_wmma_f32_16x16x4_f32) for all GEMMs: the layer is scatter/
// atomic-bound (~123M global f32 atomics), so fp32 matrix math is free and
// matches reference numerics. Gathers are L2-resident (h = 82MB << 192MB L2).
// 4 waves per workgroup share one 16-row tile (waves split the 8 N-tiles),
// keeping LDS/workgroup at ~35KB while quadrupling resident waves per WGP
// to hide atomic/gather latency.
// ---------------------------------------------------------------------------

#define H       128
#define NA      20000
#define NB      60000
#define BATCH   8

#define MI_LD   260   // >=257, LD % 64banks == 4 -> conflict-free A reads, 16B aligned
#define T_LD    132   // 128 cols, same property

typedef __attribute__((ext_vector_type(2))) float v2f;
typedef __attribute__((ext_vector_type(8))) float v8f;

__device__ __forceinline__ v8f wmma_f32_k4(v2f a, v2f b, v8f c) {
  // D(16x16,f32) = A(16x4,f32) * B(4x16,f32) + C
  return __builtin_amdgcn_wmma_f32_16x16x4_f32(
      /*neg_a=*/false, a, /*neg_b=*/false, b,
      /*c_mod=*/(short)0, c, /*reuse_a=*/false, /*reuse_b=*/false);
}

__device__ __forceinline__ float silu(float v) {
  return v * (1.0f / (1.0f + __expf(-v)));
}

// ---------------------------------------------------------------------------
__global__ void zero_kernel(float* p, size_t n) {
  size_t i = (size_t)blockIdx.x * blockDim.x + threadIdx.x;
  size_t stride = (size_t)gridDim.x * blockDim.x;
  for (; i < n; i += stride) p[i] = 0.0f;
}

__global__ void count_kernel(const long long* __restrict__ bonds, float* counts) {
  int e = blockIdx.x * blockDim.x + threadIdx.x;
  if (e >= NB) return;
  atomicAdd(&counts[(int)bonds[2 * e + 0]], 1.0f);
  atomicAdd(&counts[(int)bonds[2 * e + 1]], 1.0f);
}

// ---------------------------------------------------------------------------
// 4 waves (128 threads) cooperatively handle 16 bonds of one batch element.
// Wave w computes N-tiles {2w, 2w+1} of every GEMM.
__global__ __launch_bounds__(128) void bond_kernel(
    const float* __restrict__ h, const float* __restrict__ x,
    const long long* __restrict__ bonds,
    const float* __restrict__ Wm1, const float* __restrict__ bm1,
    const float* __restrict__ Wm2, const float* __restrict__ bm2,
    const float* __restrict__ Wc1, const float* __restrict__ bc1,
    const float* __restrict__ Wc2,
    float* __restrict__ agg, float* __restrict__ cupd)
{
  __shared__ float mi[16 * MI_LD];   // [h_src | h_dst] tile, K=0..255
  __shared__ float tA[16 * T_LD];    // m1, later t (coord head hidden)
  __shared__ float msg[16 * T_LD];   // messages
  __shared__ float dist_s[16];
  __shared__ float cd[16 * 3];
  __shared__ int   sidx[16], didx[16];
  __shared__ float cw[16];

  const int tid  = threadIdx.x;
  const int lane = tid & 31;
  const int wid  = tid >> 5;
  const int b    = blockIdx.y;
  const int e0   = blockIdx.x * 16;

  // warm L2 for the (tiny, universally reused) weight matrices
  __builtin_prefetch(Wm1 + (size_t)tid * 16, 0, 3);
  __builtin_prefetch(Wm2 + (size_t)tid * 16, 0, 3);
  __builtin_prefetch(Wc1 + (size_t)tid * 16, 0, 3);

  // -- geometry + indices (threads 0..15, one bond each) ---------------------
  if (tid < 16) {
    int e = e0 + tid;
    int s = (int)bonds[2 * e + 0];
    int d = (int)bonds[2 * e + 1];
    sidx[tid] = s; didx[tid] = d;
    const float* xs = x + ((size_t)b * NA + s) * 3;
    const float* xd = x + ((size_t)b * NA + d) * 3;
    float dx = xd[0] - xs[0], dy = xd[1] - xs[1], dz = xd[2] - xs[2];
    cd[tid * 3 + 0] = dx; cd[tid * 3 + 1] = dy; cd[tid * 3 + 2] = dz;
    dist_s[tid] = sqrtf(dx * dx + dy * dy + dz * dz);
  }
  __syncthreads();

  // -- gather mi = [h_src | h_dst] via float4 (L2-resident) ------------------
  for (int i = tid; i < 16 * 64; i += 128) {          // 16 rows x 64 float4
    int m   = i >> 6;
    int col = (i & 63) * 4;
    int atom = (col < H) ? sidx[m] : didx[m];
    float4 v = *(const float4*)(h + ((size_t)b * NA + atom) * H + (col & (H - 1)));
    float* dp = &mi[m * MI_LD + col];
    dp[0] = v.x; dp[1] = v.y; dp[2] = v.z; dp[3] = v.w;
  }
  __syncthreads();

  const int rowA   = lane & 15;          // A operand: row per lane
  const int khalf  = (lane >> 4) * 2;    // K sub-columns 0/1 vs 2/3
  const int colN   = lane & 15;          // C/D column within N-tile
  const int rowOff = (lane >> 4) * 8;    // C/D: VGPR r -> row r / r+8

  // -- GEMM1: m1 = silu(mi @ Wm1[0:256] + dist*Wm1[256] + bm1) --------------
  for (int nn = 0; nn < 2; ++nn) {
    int col = (wid * 2 + nn) * 16 + colN;
    v8f acc;
    float bias = bm1[col];
    #pragma unroll
    for (int r = 0; r < 8; ++r) acc[r] = bias;
    for (int k = 0; k < 64; ++k) {
      int k4 = k * 4;
      v2f a, bb;
      a.x  = mi[rowA * MI_LD + k4 + khalf];
      a.y  = mi[rowA * MI_LD + k4 + khalf + 1];
      bb.x = Wm1[(size_t)(k4 + khalf) * H + col];
      bb.y = Wm1[(size_t)(k4 + khalf + 1) * H + col];
      acc = wmma_f32_k4(a, bb, acc);
    }
    float wlast = Wm1[(size_t)256 * H + col];   // K=256 row: dist feature
    #pragma unroll
    for (int r = 0; r < 8; ++r) acc[r] += dist_s[r + rowOff] * wlast;
    #pragma unroll
    for (int r = 0; r < 8; ++r) tA[(r + rowOff) * T_LD + col] = silu(acc[r]);
  }
  __syncthreads();

  // -- GEMM2: messages = silu(m1 @ Wm2 + bm2) --------------------------------
  for (int nn = 0; nn < 2; ++nn) {
    int col = (wid * 2 + nn) * 16 + colN;
    v8f acc;
    float bias = bm2[col];
    #pragma unroll
    for (int r = 0; r < 8; ++r) acc[r] = bias;
    for (int k = 0; k < 32; ++k) {
      int k4 = k * 4;
      v2f a, bb;
      a.x  = tA[rowA * T_LD + k4 + khalf];
      a.y  = tA[rowA * T_LD + k4 + khalf + 1];
      bb.x = Wm2[(size_t)(k4 + khalf) * H + col];
      bb.y = Wm2[(size_t)(k4 + khalf + 1) * H + col];
      acc = wmma_f32_k4(a, bb, acc);
    }
    #pragma unroll
    for (int r = 0; r < 8; ++r) msg[(r + rowOff) * T_LD + col] = silu(acc[r]);
  }
  __syncthreads();   // msg complete; also covers WAR: GEMM2 tA reads vs GEMM3 tA writes

  // -- scatter messages into aggregated (rows split across waves) ------------
  #pragma unroll
  for (int mm = 0; mm < 4; ++mm) {
    int m = wid * 4 + mm;
    float* aS = agg + ((size_t)b * NA + sidx[m]) * H;
    float* aD = agg + ((size_t)b * NA + didx[m]) * H;
    #pragma unroll
    for (int j = 0; j < 4; ++j) {
      int n = lane + 32 * j;
      float v = msg[m * T_LD + n];
      atomicAdd(aD + n, v);
      atomicAdd(aS + n, v);
    }
  }

  // -- GEMM3: t = silu(messages @ Wc1 + bc1) ---------------------------------
  for (int nn = 0; nn < 2; ++nn) {
    int col = (wid * 2 + nn) * 16 + colN;
    v8f acc;
    float bias = bc1[col];
    #pragma unroll
    for (int r = 0; r < 8; ++r) acc[r] = bias;
    for (int k = 0; k < 32; ++k) {
      int k4 = k * 4;
      v2f a, bb;
      a.x  = msg[rowA * T_LD + k4 + khalf];
      a.y  = msg[rowA * T_LD + k4 + khalf + 1];
      bb.x = Wc1[(size_t)(k4 + khalf) * H + col];
      bb.y = Wc1[(size_t)(k4 + khalf + 1) * H + col];
      acc = wmma_f32_k4(a, bb, acc);
    }
    #pragma unroll
    for (int r = 0; r < 8; ++r) tA[(r + rowOff) * T_LD + col] = silu(acc[r]);
  }
  __syncthreads();

  // -- coord weights: cw[m] = t[m,:] . Wc2 (4 rows per wave) -----------------
  #pragma unroll
  for (int mm = 0; mm < 4; ++mm) {
    int m = wid * 4 + mm;
    float p = 0.0f;
    #pragma unroll
    for (int j = 0; j < 4; ++j) {
      int n = lane + 32 * j;
      p += tA[m * T_LD + n] * Wc2[n];
    }
    #pragma unroll
    for (int off = 16; off > 0; off >>= 1) p += __shfl_xor(p, off, 32);
    if (lane == 0) cw[m] = p;
  }
  __syncthreads();

  // -- coord scatter (+upd to dst, -upd to src) ------------------------------
  if (tid < 16) {
    int m = tid;
    float inv = 1.0f / (dist_s[m] + 1e-8f);
    float w = cw[m] * inv;
    float ux = cd[m * 3 + 0] * w, uy = cd[m * 3 + 1] * w, uz = cd[m * 3 + 2] * w;
    float* cD = cupd + ((size_t)b * NA + didx[m]) * 3;
    float* cS = cupd + ((size_t)b * NA + sidx[m]) * 3;
    atomicAdd(cD + 0,  ux); atomicAdd(cD + 1,  uy); atomicAdd(cD + 2,  uz);
    atomicAdd(cS + 0, -ux); atomicAdd(cS + 1, -uy); atomicAdd(cS + 2, -uz);
  }
}

// ---------------------------------------------------------------------------
// 4 waves handle 16 atoms: h_out = h + silu([h|agg] @ Wn1 + bn1) @ Wn2 + bn2
__global__ __launch_bounds__(128) void node_kernel(
    const float* __restrict__ h, const float* __restrict__ agg,
    const float* __restrict__ Wn1, const float* __restrict__ bn1,
    const float* __restrict__ Wn2, const float* __restrict__ bn2,
    float* __restrict__ hout)
{
  __shared__ float mi[16 * MI_LD];  // K = 256 = [h | agg]
  __shared__ float t1[16 * T_LD];

  const int tid  = threadIdx.x;
  const int lane = tid & 31;
  const int wid  = tid >> 5;
  const int b    = blockIdx.y;
  const int a0   = blockIdx.x * 16;

  __builtin_prefetch(Wn1 + (size_t)tid * 16, 0, 3);
  __builtin_prefetch(Wn2 + (size_t)tid * 16, 0, 3);

  for (int i = tid; i < 16 * 64; i += 128) {
    int m   = i >> 6;
    int col = (i & 63) * 4;
    const float* sp = (col < H)
        ? (h   + ((size_t)b * NA + a0 + m) * H + col)
        : (agg + ((size_t)b * NA + a0 + m) * H + (col - H));
    float4 v = *(const float4*)sp;
    float* dp = &mi[m * MI_LD + col];
    dp[0] = v.x; dp[1] = v.y; dp[2] = v.z; dp[3] = v.w;
  }
  __syncthreads();

  const int rowA   = lane & 15;
  const int khalf  = (lane >> 4) * 2;
  const int colN   = lane & 15;
  const int rowOff = (lane >> 4) * 8;

  for (int nn = 0; nn < 2; ++nn) {
    int col = (wid * 2 + nn) * 16 + colN;
    v8f acc;
    float bias = bn1[col];
    #pragma unroll
    for (int r = 0; r < 8; ++r) acc[r] = bias;
    for (int k = 0; k < 64; ++k) {          // K = 256
      int k4 = k * 4;
      v2f a, bb;
      a.x  = mi[rowA * MI_LD + k4 + khalf];
      a.y  = mi[rowA * MI_LD + k4 + khalf + 1];
      bb.x = Wn1[(size_t)(k4 + khalf) * H + col];
      bb.y = Wn1[(size_t)(k4 + khalf + 1) * H + col];
      acc = wmma_f32_k4(a, bb, acc);
    }
    #pragma unroll
    for (int r = 0; r < 8; ++r) t1[(r + rowOff) * T_LD + col] = silu(acc[r]);
  }
  __syncthreads();

  for (int nn = 0; nn < 2; ++nn) {
    int col = (wid * 2 + nn) * 16 + colN;
    v8f acc;
    float bias = bn2[col];
    #pragma unroll
    for (int r = 0; r < 8; ++r) acc[r] = bias;
    for (int k = 0; k < 32; ++k) {          // K = 128
      int k4 = k * 4;
      v2f a, bb;
      a.x  = t1[rowA * T_LD + k4 + khalf];
      a.y  = t1[rowA * T_LD + k4 + khalf + 1];
      bb.x = Wn2[(size_t)(k4 + khalf) * H + col];
      bb.y = Wn2[(size_t)(k4 + khalf + 1) * H + col];
      acc = wmma_f32_k4(a, bb, acc);
    }
    #pragma unroll
    for (int r = 0; r < 8; ++r) {
      size_t idx = ((size_t)b * NA + a0 + r + rowOff) * H + col;
      hout[idx] = h[idx] + acc[r];          // residual
    }
  }
}

// ---------------------------------------------------------------------------
__global__ void x_kernel(const float* __restrict__ x, const float* __restrict__ cupd,
                         const float* __restrict__ counts, float* __restrict__ xout)
{
  int i = blockIdx.x * blockDim.x + threadIdx.x;
  if (i >= BATCH * NA * 3) return;
  int atom = (i / 3) % NA;
  float c = counts[atom];
  c = (c < 1.0f) ? 1.0f : c;
  xout[i] = x[i] + cupd[i] / c;
}

// ---------------------------------------------------------------------------
extern "C" void kernel_launch(void* const* d_in, const int* in_sizes, int n_in,
                              void* d_out, int out_size, void* d_ws, size_t ws_size,
                              hipStream_t stream) {
  const float*      h     = (const float*)d_in[0];
  const float*      x     = (const float*)d_in[1];
  const long long*  bonds = (const long long*)d_in[2];   // int64 per reference
  const float* Wm1 = (const float*)d_in[3];
  const float* bm1 = (const float*)d_in[4];
  const float* Wm2 = (const float*)d_in[5];
  const float* bm2 = (const float*)d_in[6];
  const float* Wn1 = (const float*)d_in[7];
  const float* bn1 = (const float*)d_in[8];
  const float* Wn2 = (const float*)d_in[9];
  const float* bn2 = (const float*)d_in[10];
  const float* Wc1 = (const float*)d_in[11];
  const float* bc1 = (const float*)d_in[12];
  const float* Wc2 = (const float*)d_in[13];

  // Workspace layout (~84 MB): aggregated | coord_update | bond_counts
  float* agg    = (float*)d_ws;
  float* cupd   = agg  + (size_t)BATCH * NA * H;
  float* counts = cupd + (size_t)BATCH * NA * 3;
  size_t ws_floats = (size_t)BATCH * NA * H + (size_t)BATCH * NA * 3 + NA;

  float* hout = (float*)d_out;                       // [8,20000,128]
  float* xout = hout + (size_t)BATCH * NA * H;       // [8,20000,3]

  zero_kernel<<<(int)((ws_floats + 1023) / 1024), 1024, 0, stream>>>((float*)d_ws, ws_floats);
  count_kernel<<<(NB + 255) / 256, 256, 0, stream>>>(bonds, counts);

  bond_kernel<<<dim3(NB / 16, BATCH), 128, 0, stream>>>(
      h, x, bonds, Wm1, bm1, Wm2, bm2, Wc1, bc1, Wc2, agg, cupd);

  node_kernel<<<dim3(NA / 16, BATCH), 128, 0, stream>>>(
      h, agg, Wn1, bn1, Wn2, bn2, hout);

  x_kernel<<<(BATCH * NA * 3 + 255) / 256, 256, 0, stream>>>(x, cupd, counts, xout);
}